// DisRNN_13314398617615
// MI455X (gfx1250) — compile-verified
//
#include <hip/hip_runtime.h>
#include <hip/hip_bf16.h>

typedef float v2f __attribute__((ext_vector_type(2)));
typedef float v8f __attribute__((ext_vector_type(8)));

#define BZ   64     // Z latents
#define BD   66     // D = Z + NOBS
#define BH   16     // H1 = H2 = 16
#define BT   16     // batch rows per workgroup

// out layout (concatenated flat): y[B,2] | z_tilde[B,64] | kld_g[B] | kld_u[B,64]
#define OFF_ZT   (16384 * 2)
#define OFF_KLDG (OFF_ZT + 16384 * 64)
#define OFF_KLDU (OFF_KLDG + 16384)

// gfx1250 async global->LDS copy: each lane moves one dword from its own
// global address to its own LDS byte offset (tracked by ASYNCcnt).
__device__ __forceinline__ void async_copy_b32(const float* gp, float* lp) {
    unsigned       lofs = (unsigned)(unsigned long long)(uintptr_t)lp; // LDS aperture: addr[31:0]
    unsigned long long ga = (unsigned long long)(uintptr_t)gp;
    asm volatile("global_load_async_to_lds_b32 %0, %1, off"
                 :: "v"(lofs), "v"(ga) : "memory");
}
__device__ __forceinline__ void wait_async0() {
    asm volatile("s_wait_asynccnt 0x0" ::: "memory");
}

__global__ __launch_bounds__(256) void disrnn_fused(
    const float* __restrict__ latents, const float* __restrict__ obs,
    const float* __restrict__ noise_u, const float* __restrict__ noise_g,
    const float* __restrict__ z_0,     const float* __restrict__ mult_u,
    const float* __restrict__ logvar_u,const float* __restrict__ mult_g,
    const float* __restrict__ logvar_g,
    const float* __restrict__ W1, const float* __restrict__ b1,
    const float* __restrict__ W2, const float* __restrict__ b2,
    const float* __restrict__ W3, const float* __restrict__ b3,
    const float* __restrict__ Wc1, const float* __restrict__ bc1,
    const float* __restrict__ Wc2, const float* __restrict__ bc2,
    const float* __restrict__ Wc3, const float* __restrict__ bc3,
    const int* __restrict__ t0p, float* __restrict__ out)
{
    // ---- LDS (~125 KB; ISA allows up to 320 KB/WG) ----
    __shared__ float xt_s[2][64 * 144]; // raw noise chunk, dbl-buffered: [z][b(stride 9)][dd]
    __shared__ float sv_s[BD * 64];     // sqrt(var_u) = exp(logvar_u/2)   [d][z]
    __shared__ float mu_s[BD * 64];     // mult_u                          [d][z]
    __shared__ float xs_s[16 * 68];     // x = concat(lat, obs) tile       [b][d]
    __shared__ float nl_s[16 * 65];     // new_latents [b][z]
    __shared__ float zt_s[16 * 65];     // z_tilde     [b][z]
    __shared__ float hc1_s[16 * 33];
    __shared__ float hc2_s[16 * 33];
    __shared__ float kldg_s[16];
    __shared__ float kldc0_s[64];

    const int t     = threadIdx.x;
    const int lane  = t & 31;
    const int wave  = t >> 5;
    const int bbase = blockIdx.x * BT;
    const int m16   = lane & 15;
    const int half  = lane >> 4;

    float* outY    = out;
    float* outZT   = out + OFF_ZT;
    float* outKLDG = out + OFF_KLDG;
    float* outKLDU = out + OFF_KLDU;

    // ---- stage x tile, coefficient tables, kld constants ----
    const int tz = *t0p;
    for (int flat = t; flat < 16 * 68; flat += 256) {
        int d = flat % 68, b = flat / 68;
        float v = 0.0f;
        if (d < 64)       v = tz ? z_0[d] : latents[(bbase + b) * 64 + d];
        else if (d < 66)  v = obs[(bbase + b) * 2 + (d - 64)];
        xs_s[b * 68 + d] = v;
    }
    for (int i = t; i < BD * 64; i += 256) {       // flat = d*64 + z
        sv_s[i] = __expf(0.5f * logvar_u[i]);      // sqrt(exp(lv)) == exp(lv/2)
        mu_s[i] = mult_u[i];
    }
    if (t < 64) {
        float s = 0.0f;
        for (int d = 0; d < BD; ++d) {
            float lv = logvar_u[d * 64 + t];
            s += 1.0f + lv - __expf(lv);
        }
        kldc0_s[t] = -0.5f * s;
    }
    if (t < 16) kldg_s[t] = 0.0f;
    __syncthreads();

    // ---- kick off chunk 0 DMA, overlap with kld_u math ----
    // async-issue of raw noise chunk c (d-chunks of 8, d clamped so every
    // transfer is in-bounds; invalid tail lanes masked at consume time)
    #define ISSUE_CHUNK(c)                                                     \
    {                                                                          \
        const int d0_ = (c) * 8;                                               \
        float* xb_ = xt_s[(c) & 1];                                            \
        _Pragma("unroll")                                                      \
        for (int i = 0; i < 32; ++i) {                                         \
            int flat = i * 256 + t;                                            \
            int z = flat & 63, bd = flat >> 6, dd = bd & 7, b = bd >> 3;       \
            int d  = d0_ + dd;                                                 \
            int dc = d < BD ? d : BD - 1;                                      \
            async_copy_b32(&noise_u[((bbase + b) * BD + dc) * 64 + z],         \
                           &xb_[z * 144 + b * 9 + dd]);                        \
        }                                                                      \
    }

    ISSUE_CHUNK(0)

    // kld_u[b,z] = kldc0[z] + 0.5 * sum_d (x[b,d]*mult_u[d,z])^2  (VALU, overlaps DMA)
    #pragma unroll
    for (int i = 0; i < 4; ++i) {
        int flat = i * 256 + t;
        int z = flat & 63, b = flat >> 6;
        float acc = 0.0f;
        for (int d = 0; d < BD; ++d) {
            float m = xs_s[b * 68 + d] * mu_s[d * 64 + z];
            acc += m * m;
        }
        outKLDU[(bbase + b) * 64 + z] = kldc0_s[z] + 0.5f * acc;
    }
    wait_async0();
    __syncthreads();

    // ---- layer-1 grouped GEMM, software-pipelined over 9 d-chunks ----
    v8f acc1[8];
    #pragma unroll
    for (int zi = 0; zi < 8; ++zi) acc1[zi] = (v8f){0,0,0,0,0,0,0,0};

    for (int c = 0; c < 9; ++c) {
        if (c + 1 < 9) ISSUE_CHUNK(c + 1)          // DMA next chunk into other buffer
        const int d0 = c * 8;
        const float* xb = xt_s[c & 1];
        #pragma unroll
        for (int zi = 0; zi < 8; ++zi) {
            const int z = wave * 8 + zi;
            const float* xz = &xb[z * 144];
            #pragma unroll
            for (int k0 = 0; k0 < 8; k0 += 4) {
                int kk  = k0 + half * 2;
                int d   = d0 + kk;
                int dc0 = d     < BD ? d     : BD - 1;
                int dc1 = d + 1 < BD ? d + 1 : BD - 1;
                // x_tilde built on the fly: raw*sv + x*mu ; masked past d=65
                float r0 = xz[m16 * 9 + kk], r1 = xz[m16 * 9 + kk + 1];
                float t0v = r0 * sv_s[dc0 * 64 + z] + xs_s[m16 * 68 + dc0] * mu_s[dc0 * 64 + z];
                float t1v = r1 * sv_s[dc1 * 64 + z] + xs_s[m16 * 68 + dc1] * mu_s[dc1 * 64 + z];
                v2f a;
                a.x = (d     < BD) ? t0v : 0.0f;
                a.y = (d + 1 < BD) ? t1v : 0.0f;
                v2f bb;                            // clamped loads, A==0 kills tail
                bb.x = W1[(z * BD + dc0) * BH + m16];
                bb.y = W1[(z * BD + dc1) * BH + m16];
                acc1[zi] = __builtin_amdgcn_wmma_f32_16x16x4_f32(
                    false, a, false, bb, (short)0, acc1[zi], false, false);
            }
        }
        wait_async0();
        __syncthreads();
    }

    // ---- layers 2/3 per latent (wave-private LDS slab aliased into retired buffer 0) ----
    float* hs = &xt_s[0][wave * 272];              // [16 rows][17]
    #pragma unroll
    for (int zi = 0; zi < 8; ++zi) {
        const int z = wave * 8 + zi;
        float b1v = b1[z * BH + m16];
        #pragma unroll
        for (int v = 0; v < 8; ++v) {
            int row = v + 8 * half;
            hs[row * 17 + m16] = fmaxf(acc1[zi][v] + b1v, 0.0f);
        }
        v8f acc2 = (v8f){0,0,0,0,0,0,0,0};
        #pragma unroll
        for (int k0 = 0; k0 < 16; k0 += 4) {
            int kk = k0 + half * 2;
            v2f a; a.x = hs[m16 * 17 + kk]; a.y = hs[m16 * 17 + kk + 1];
            v2f bb;
            bb.x = W2[(z * BH + kk    ) * BH + m16];
            bb.y = W2[(z * BH + kk + 1) * BH + m16];
            acc2 = __builtin_amdgcn_wmma_f32_16x16x4_f32(
                false, a, false, bb, (short)0, acc2, false, false);
        }
        float b2v = b2[z * BH + m16];
        #pragma unroll
        for (int v = 0; v < 8; ++v) {
            int row = v + 8 * half;
            hs[row * 17 + m16] = fmaxf(acc2[v] + b2v, 0.0f);
        }
        // o[row, col] = h2 @ W3[z] + b3 ; col = half in {0,1}
        float o = b3[z * 2 + half];
        #pragma unroll
        for (int k = 0; k < 16; ++k)
            o += hs[m16 * 17 + k] * W3[(z * BH + k) * 2 + half];
        if (half == 1) hs[m16 * 17 + 16] = o;      // stash w-logit in pad column
        if (half == 0) {
            float ow   = hs[m16 * 17 + 16];
            float w    = 1.0f / (1.0f + __expf(-ow));
            float latv = xs_s[m16 * 68 + z];
            nl_s[m16 * 65 + z] = (1.0f - w) * latv + o * w;
        }
    }
    __syncthreads();

    // ---- global bottleneck: z_tilde, kld_g ----
    {
        int b = t >> 4, part = t & 15;
        float partial = 0.0f;
        #pragma unroll
        for (int j = 0; j < 4; ++j) {
            int z = part * 4 + j;
            float lg = logvar_g[z];
            float vg = __expf(lg);
            float m  = mult_g[z] * nl_s[b * 65 + z];
            float zt = noise_g[(bbase + b) * 64 + z] * __expf(0.5f * lg) + m;
            zt_s[b * 65 + z] = zt;
            outZT[(bbase + b) * 64 + z] = zt;
            partial += 1.0f + lg - m * m - vg;
        }
        atomicAdd(&kldg_s[b], partial);
    }
    __syncthreads();
    if (t < 16) outKLDG[bbase + t] = -0.5f * kldg_s[t];

    // ---- choice MLP: 16 rows, 64->32->32->2 (VALU, tiny) ----
    {
        int b = t >> 4, p = t & 15;
        #pragma unroll
        for (int it = 0; it < 2; ++it) {
            int col = p + it * 16;
            float a = bc1[col];
            for (int k = 0; k < 64; ++k) a += zt_s[b * 65 + k] * Wc1[k * 32 + col];
            hc1_s[b * 33 + col] = fmaxf(a, 0.0f);
        }
    }
    __syncthreads();
    {
        int b = t >> 4, p = t & 15;
        #pragma unroll
        for (int it = 0; it < 2; ++it) {
            int col = p + it * 16;
            float a = bc2[col];
            for (int k = 0; k < 32; ++k) a += hc1_s[b * 33 + k] * Wc2[k * 32 + col];
            hc2_s[b * 33 + col] = fmaxf(a, 0.0f);
        }
    }
    __syncthreads();
    if (t < 32) {
        int b = t >> 1, col = t & 1;
        float a = bc3[col];
        for (int k = 0; k < 32; ++k) a += hc2_s[b * 33 + k] * Wc3[k * 2 + col];
        outY[(bbase + b) * 2 + col] = a;
    }
}

extern "C" void kernel_launch(void* const* d_in, const int* in_sizes, int n_in,
                              void* d_out, int out_size, void* d_ws, size_t ws_size,
                              hipStream_t stream) {
    const float* latents  = (const float*)d_in[0];
    const float* obs      = (const float*)d_in[1];
    const float* noise_u  = (const float*)d_in[2];
    const float* noise_g  = (const float*)d_in[3];
    const float* z_0      = (const float*)d_in[4];
    const float* mult_u   = (const float*)d_in[5];
    const float* logvar_u = (const float*)d_in[6];
    const float* mult_g   = (const float*)d_in[7];
    const float* logvar_g = (const float*)d_in[8];
    const float* W1  = (const float*)d_in[9];
    const float* b1  = (const float*)d_in[10];
    const float* W2  = (const float*)d_in[11];
    const float* b2  = (const float*)d_in[12];
    const float* W3  = (const float*)d_in[13];
    const float* b3  = (const float*)d_in[14];
    const float* Wc1 = (const float*)d_in[15];
    const float* bc1 = (const float*)d_in[16];
    const float* Wc2 = (const float*)d_in[17];
    const float* bc2 = (const float*)d_in[18];
    const float* Wc3 = (const float*)d_in[19];
    const float* bc3 = (const float*)d_in[20];
    const int*   t0p = (const int*)d_in[21];
    float* out = (float*)d_out;

    const int B = in_sizes[0] / 64;          // 16384
    dim3 grid(B / BT), block(256);
    disrnn_fused<<<grid, block, 0, stream>>>(
        latents, obs, noise_u, noise_g, z_0, mult_u, logvar_u, mult_g, logvar_g,
        W1, b1, W2, b2, W3, b3, Wc1, bc1, Wc2, bc2, Wc3, bc3, t0p, out);
}